// HardLogLikelihood_61314953118190
// MI455X (gfx1250) — compile-verified
//
#include <hip/hip_runtime.h>
#include <math.h>

typedef __attribute__((ext_vector_type(2))) float v2f;
typedef __attribute__((ext_vector_type(8))) float v8f;

#define LOG_QUARTER (-1.3862943611198906f)
#define FMAXV 3.4028234663852886e38f   /* finfo(float32).max */

// out[e][k] = log(0.25) + logdr[e][k>>2] + logdi[e][k&3]
// D = A1*B1 + (A2*B2 + C): two chained V_WMMA_F32_16X16X4_F32 per wave.
// Lane pairing: lanes m and m+16 both process tile row m; lower half computes
// axis terms k=0,1, upper half k=2,3 -> A operands are produced directly in
// the documented 16x4 f32 A-matrix VGPR layout (no cross-lane traffic).
// Requires the grid to cover a multiple of 16 elements per wave (branch-free,
// EXEC all-ones, as WMMA requires); remainder handled by a scalar tail kernel.
__global__ __launch_bounds__(256) void qam16_hard_llr_wmma(
    const float* __restrict__ xr, const float* __restrict__ xi,
    const float* __restrict__ snrp, const float* __restrict__ levels,
    const float* __restrict__ pre, const float* __restrict__ pim,
    float* __restrict__ out)
{
    const int lane   = threadIdx.x & 31;
    const int m      = lane & 15;                          // tile row / column
    const bool hiHalf = (lane >= 16);
    const int kb     = hiHalf ? 2 : 0;                     // this lane's K base
    const int waveId = (blockIdx.x << 3) + (threadIdx.x >> 5);
    const int e      = (waveId << 4) + m;                  // element for row m

    // ---- per-lane element math (VALU / transcendental pipe) ----
    const float x = xr[e];
    const float y = xi[e];
    const float s = sqrtf(snrp[e]);

    const float l0 = levels[0], l1 = levels[1], l2 = levels[2], l3 = levels[3];
    const float m0 = 0.5f * (l0 + l1);
    const float m1 = 0.5f * (l1 + l2);
    const float m2 = 0.5f * (l2 + l3);

    // this lane's two candidate amplitudes per axis
    // (points_real[k]=levels[k>>2], points_imag[k]=levels[k&3])
    const float cr0 = pre[4 * (kb + 0)], cr1 = pre[4 * (kb + 1)];
    const float ci0 = pim[kb + 0],       ci1 = pim[kb + 1];

    // searchsorted(mid, x) side='left': left edge = largest mid < x (else -max),
    // right edge = smallest mid >= x (else +max)
    float lr = -FMAXV; lr = (x > m0) ? m0 : lr; lr = (x > m1) ? m1 : lr; lr = (x > m2) ? m2 : lr;
    float rr =  FMAXV; rr = (m2 >= x) ? m2 : rr; rr = (m1 >= x) ? m1 : rr; rr = (m0 >= x) ? m0 : rr;
    float li = -FMAXV; li = (y > m0) ? m0 : li; li = (y > m1) ? m1 : li; li = (y > m2) ? m2 : li;
    float ri =  FMAXV; ri = (m2 >= y) ? m2 : ri; ri = (m1 >= y) ? m1 : ri; ri = (m0 >= y) ? m0 : ri;

    // A-matrix layout (16x4 f32, 2 VGPRs): lanes 0-15 hold K=0 (v0), K=1 (v1);
    // lanes 16-31 hold K=2 (v0), K=3 (v1); row M = lane&15. Natively produced.
    v2f A1, A2;
    A1.x = logf(erff((rr - cr0) * s) - erff((lr - cr0) * s));
    A1.y = logf(erff((rr - cr1) * s) - erff((lr - cr1) * s));
    A2.x = logf(erff((ri - ci0) * s) - erff((li - ci0) * s));
    A2.y = logf(erff((ri - ci1) * s) - erff((li - ci1) * s));

    // constant 0/1 selection matrices in B layout (4x16 f32, 2 VGPRs):
    // VGPR v, lanes 0-15 -> row K=v, N=lane; lanes 16-31 -> row K=v+2.
    v2f B1, B2;
    B1.x = ((m >> 2) == (kb + 0)) ? 1.0f : 0.0f;
    B1.y = ((m >> 2) == (kb + 1)) ? 1.0f : 0.0f;
    B2.x = ((m & 3)  == (kb + 0)) ? 1.0f : 0.0f;
    B2.y = ((m & 3)  == (kb + 1)) ? 1.0f : 0.0f;

    v8f acc;
#pragma unroll
    for (int r = 0; r < 8; ++r) acc[r] = LOG_QUARTER;

    // D = A1*B1 + (A2*B2 + C)  (matrix pipe, co-executes with VALU)
    acc = __builtin_amdgcn_wmma_f32_16x16x4_f32(
              false, A2, false, B2, (short)0, acc, false, false);
    acc = __builtin_amdgcn_wmma_f32_16x16x4_f32(
              false, A1, false, B1, (short)0, acc, false, false);

    // D layout: VGPR r, lanes 0-15 -> row r, col lane; lanes 16-31 -> row r+8.
    // One base address per lane; 8 stores with immediate offsets r*64B.
    const int mrow = (waveId << 4) + (hiHalf ? 8 : 0);
    float* __restrict__ p = out + ((size_t)mrow << 4) + m;
#pragma unroll
    for (int r = 0; r < 8; ++r) p[(size_t)r * 16] = acc[r];
}

// Scalar remainder kernel (not launched when E % 128 == 0, as here).
__global__ __launch_bounds__(64) void qam16_hard_llr_tail(
    const float* __restrict__ xr, const float* __restrict__ xi,
    const float* __restrict__ snrp, const float* __restrict__ levels,
    const float* __restrict__ pre, const float* __restrict__ pim,
    float* __restrict__ out, int start, int E)
{
    const int e = start + blockIdx.x * blockDim.x + threadIdx.x;
    if (e >= E) return;

    const float x = xr[e];
    const float y = xi[e];
    const float s = sqrtf(snrp[e]);

    const float l0 = levels[0], l1 = levels[1], l2 = levels[2], l3 = levels[3];
    const float m0 = 0.5f * (l0 + l1);
    const float m1 = 0.5f * (l1 + l2);
    const float m2 = 0.5f * (l2 + l3);

    float lr = -FMAXV; lr = (x > m0) ? m0 : lr; lr = (x > m1) ? m1 : lr; lr = (x > m2) ? m2 : lr;
    float rr =  FMAXV; rr = (m2 >= x) ? m2 : rr; rr = (m1 >= x) ? m1 : rr; rr = (m0 >= x) ? m0 : rr;
    float li = -FMAXV; li = (y > m0) ? m0 : li; li = (y > m1) ? m1 : li; li = (y > m2) ? m2 : li;
    float ri =  FMAXV; ri = (m2 >= y) ? m2 : ri; ri = (m1 >= y) ? m1 : ri; ri = (m0 >= y) ? m0 : ri;

    float ldr[4], ldi[4];
#pragma unroll
    for (int i = 0; i < 4; ++i) {
        const float cr = pre[4 * i];
        const float ci = pim[i];
        ldr[i] = logf(erff((rr - cr) * s) - erff((lr - cr) * s));
        ldi[i] = logf(erff((ri - ci) * s) - erff((li - ci) * s));
    }
#pragma unroll
    for (int i = 0; i < 4; ++i)
#pragma unroll
        for (int j = 0; j < 4; ++j)
            out[(size_t)e * 16 + i * 4 + j] = LOG_QUARTER + ldr[i] + ldi[j];
}

extern "C" void kernel_launch(void* const* d_in, const int* in_sizes, int n_in,
                              void* d_out, int out_size, void* d_ws, size_t ws_size,
                              hipStream_t stream) {
    const float* xr     = (const float*)d_in[0];
    const float* xi     = (const float*)d_in[1];
    const float* snr    = (const float*)d_in[2];
    const float* levels = (const float*)d_in[3];
    const float* pre    = (const float*)d_in[4];
    const float* pim    = (const float*)d_in[5];
    float* out = (float*)d_out;

    const int E = in_sizes[0];                 // B*M*N = 2^21
    // Main kernel: 256 threads = 8 wave32s, 16 elements per wave = 128/block.
    const int fullBlocks = E / 128;            // 16384 for E = 2^21
    const int covered    = fullBlocks * 128;
    if (fullBlocks > 0) {
        hipLaunchKernelGGL(qam16_hard_llr_wmma, dim3(fullBlocks), dim3(256), 0, stream,
                           xr, xi, snr, levels, pre, pim, out);
    }
    const int rem = E - covered;               // 0 for this problem size
    if (rem > 0) {
        hipLaunchKernelGGL(qam16_hard_llr_tail, dim3((rem + 63) / 64), dim3(64), 0, stream,
                           xr, xi, snr, levels, pre, pim, out, covered, E);
    }
}